// FeatureGenKeras_2095944041142
// MI455X (gfx1250) — compile-verified
//
#include <hip/hip_runtime.h>
#include <math.h>

typedef __attribute__((ext_vector_type(2))) float v2f;
typedef __attribute__((ext_vector_type(8))) float v8f;

#define NFEAT 115
#define NXF 86
#define ROWLEN 1196
#define OUTROWS 100

// ---------------- pass 0: zero the atomic counters ----------------
__global__ void init_cnt_kernel(unsigned int* cnt) {
    if (threadIdx.x < 2) cnt[threadIdx.x] = 0u;
}

// ---------------- pass 1: streaming reduction over hand regions ----------------
// One block per frame. lefth = x[t,40:61,:] (63 floats), righth = x[t,94:115,:].
// Produces global nonzero counts (for cond) and per-frame value sums (for keep).
__global__ void count_keep_kernel(const float* __restrict__ x,
                                  unsigned int* __restrict__ cnt,
                                  unsigned char* __restrict__ keepL,
                                  unsigned char* __restrict__ keepR) {
    __shared__ float s_sl[64], s_sr[64];
    __shared__ unsigned int s_cl[64], s_cr[64];
    const int t = blockIdx.x;
    const int tid = threadIdx.x;
    const float* fx = x + (size_t)t * (NFEAT * 3);
    float sl = 0.f, sr = 0.f; unsigned int cl = 0u, cr = 0u;
    if (tid < 63) {
        float a = fx[40 * 3 + tid];
        float b = fx[94 * 3 + tid];
        a = (a != a) ? 0.f : a;   // NaN -> 0
        b = (b != b) ? 0.f : b;
        sl = a; sr = b;
        cl = (a != 0.f) ? 1u : 0u;
        cr = (b != 0.f) ? 1u : 0u;
    }
    s_sl[tid] = sl; s_sr[tid] = sr; s_cl[tid] = cl; s_cr[tid] = cr;
    __syncthreads();
    for (int off = 32; off > 0; off >>= 1) {
        if (tid < off) {
            s_sl[tid] += s_sl[tid + off];
            s_sr[tid] += s_sr[tid + off];
            s_cl[tid] += s_cl[tid + off];
            s_cr[tid] += s_cr[tid + off];
        }
        __syncthreads();
    }
    if (tid == 0) {
        keepL[t] = (s_sl[0] != 0.f) ? 1 : 0;
        keepR[t] = (s_sr[0] != 0.f) ? 1 : 0;
        atomicAdd(&cnt[0], s_cl[0]);
        atomicAdd(&cnt[1], s_cr[0]);
    }
}

// ---------------- pass 2: stable-partition order[0..100] via block scan ----------------
// Stable argsort of !keep == keeps (ascending index) first, then non-keeps.
// Prefix sums over 1024 threads give both ranks without a second scan:
// falses-before(t0) = t0 - trues-before(t0).
__global__ void order_kernel(const unsigned int* __restrict__ cnt,
                             const unsigned char* __restrict__ keepL,
                             const unsigned char* __restrict__ keepR,
                             int* __restrict__ order, int T) {
    __shared__ int s[1024];
    const int tid = threadIdx.x;
    const bool cond = cnt[0] > cnt[1];
    const unsigned char* keep = cond ? keepL : keepR;
    const int CH = (T + 1023) >> 10;
    const int t0 = tid * CH;
    const int t1 = (t0 + CH < T) ? (t0 + CH) : T;
    int c = 0;
    for (int t = t0; t < t1; ++t) c += keep[t] ? 1 : 0;
    s[tid] = c;
    __syncthreads();
    for (int off = 1; off < 1024; off <<= 1) {   // Hillis-Steele inclusive scan
        int v = s[tid];
        int add = (tid >= off) ? s[tid - off] : 0;
        __syncthreads();
        s[tid] = v + add;
        __syncthreads();
    }
    const int incl = s[tid];
    const int excl = incl - c;
    const int K = s[1023];
    const int vb = (t0 < T) ? t0 : T;
    int rT = excl;              // rank among keeps
    int rF = K + (vb - excl);   // K + rank among non-keeps
    for (int t = t0; t < t1; ++t) {
        if (keep[t]) { if (rT <= OUTROWS) order[rT] = t; rT++; }
        else         { if (rF <= OUTROWS) order[rF] = t; rF++; }
    }
}

// ---------------- pass 3: gather + dxyz + WMMA pairwise distances ----------------
__device__ __forceinline__ float ptNorm(const float* A, int F, int D, int p) {
    float xx = A[(F + p) * 3 + 0];
    float yy = A[(F + p) * 3 + 1];
    float s = xx * xx + yy * yy;
    if (D > 2) { float zz = A[(F + p) * 3 + 2]; s += zz * zz; }
    return s;
}

__global__ void feat_kernel(const float* __restrict__ x,
                            const unsigned int* __restrict__ cnt,
                            const int* __restrict__ order,
                            float* __restrict__ out) {
    __shared__ float A[NXF * 3];   // xfeat[order[row]]
    __shared__ float B[NXF * 3];   // xfeat[order[row+1]] (for dxyz)
    const int row = blockIdx.x;
    const int tid = threadIdx.x;
    const bool cond = cnt[0] > cnt[1];
    const int tA = order[row];
    const int tB = order[row + 1];

    // stage permuted + cleaned + x-negated features into LDS
    for (int j = tid; j < NXF * 3; j += blockDim.x) {
        int f = j / 3, c = j - f * 3;
        int src;
        if (f < 21)      src = cond ? (40 + f) : (94 + f);  // hand
        else if (f < 46) src = f + 40;                      // pose (61 + f-21)
        else             src = f - 46;                      // lip
        float vA = x[(size_t)tA * (NFEAT * 3) + src * 3 + c];
        float vB = x[(size_t)tB * (NFEAT * 3) + src * 3 + c];
        vA = (vA != vA) ? 0.f : vA;
        vB = (vB != vB) ? 0.f : vB;
        if (cond && c == 0) { vA = -vA; vB = -vB; }
        A[j] = vA; B[j] = vB;
    }
    __syncthreads();

    float* o = out + (size_t)row * ROWLEN;

    // base features [0,153) and dxyz [153,306)
    for (int k = tid; k < 306; k += blockDim.x) {
        int kk = (k < 153) ? k : (k - 153);
        int idx;
        if (kk < 63)       idx = kk;                                       // 21 pts x 3
        else if (kk < 113) { int i2 = kk - 63;  idx = (21 + (i2 >> 1)) * 3 + (i2 & 1); } // 25 x 2
        else               { int i2 = kk - 113; idx = (46 + (i2 >> 1)) * 3 + (i2 & 1); } // 20 x 2
        o[k] = (k < 153) ? A[idx] : (A[idx] - B[idx]);
    }

    // ---- pairwise distances via V_WMMA_F32_16X16X4_F32 Gram matrices ----
    // wave w handles point-set w; d^2(m,n) = |p_m|^2 + |p_n|^2 - 2*G[m][n]
    const int w = tid >> 5;
    const int lane = tid & 31;
    const int setN[4] = {21, 25, 20, 20};
    const int setD[4] = {3, 2, 2, 2};
    const int setF[4] = {0, 21, 46, 66};
    const int setO[4] = {306, 516, 816, 1006};
    const int N = setN[w], D = setD[w], F = setF[w], OB = setO[w];
    const int lp = lane & 15;
    const bool hiL = lane >= 16;

    // f32 16x4 A-matrix layout == 4x16 B-matrix layout for a symmetric product:
    // lanes 0-15 hold (K=0,K=1)=(x,y) of point lp; lanes 16-31 hold (K=2,K=3)=(z,0).
    v2f r0 = {0.f, 0.f}, r1 = {0.f, 0.f};
    {
        int p = lp;                    // tile 0: points 0..15
        if (p < N) {
            if (!hiL)       { r0.x = A[(F + p) * 3 + 0]; r0.y = A[(F + p) * 3 + 1]; }
            else if (D > 2) { r0.x = A[(F + p) * 3 + 2]; }
        }
        p = 16 + lp;                   // tile 1: points 16..31
        if (p < N) {
            if (!hiL)       { r1.x = A[(F + p) * 3 + 0]; r1.y = A[(F + p) * 3 + 1]; }
            else if (D > 2) { r1.x = A[(F + p) * 3 + 2]; }
        }
    }

    v8f cz = {0.f, 0.f, 0.f, 0.f, 0.f, 0.f, 0.f, 0.f};
    // 2x2 tiling of the <=32x32 Gram; only upper-triangle tiles needed.
    v8f G00 = __builtin_amdgcn_wmma_f32_16x16x4_f32(false, r0, false, r0, (short)0, cz, false, false);
    v8f G01 = __builtin_amdgcn_wmma_f32_16x16x4_f32(false, r0, false, r1, (short)0, cz, false, false);
    v8f G11 = __builtin_amdgcn_wmma_f32_16x16x4_f32(false, r1, false, r1, (short)0, cz, false, false);

    // C/D layout: VGPR r, lanes 0-15 -> M=r, lanes 16-31 -> M=r+8; N = lane&15
    const int n0 = lp;
#pragma unroll
    for (int r = 0; r < 8; ++r) {
        const int m0 = r + (hiL ? 8 : 0);
        {   // tile (0,0): m=m0, n=n0
            int m = m0, n = n0;
            if (m < n && n < N) {
                float d2 = ptNorm(A, F, D, m) + ptNorm(A, F, D, n) - 2.f * G00[r];
                int pi = m * (2 * N - m - 1) / 2 + (n - m - 1);
                o[OB + pi] = sqrtf(fmaxf(d2, 0.f));
            }
        }
        {   // tile (0,1): m=m0, n=16+n0
            int m = m0, n = 16 + n0;
            if (m < n && n < N) {
                float d2 = ptNorm(A, F, D, m) + ptNorm(A, F, D, n) - 2.f * G01[r];
                int pi = m * (2 * N - m - 1) / 2 + (n - m - 1);
                o[OB + pi] = sqrtf(fmaxf(d2, 0.f));
            }
        }
        {   // tile (1,1): m=16+m0, n=16+n0
            int m = 16 + m0, n = 16 + n0;
            if (m < n && n < N) {
                float d2 = ptNorm(A, F, D, m) + ptNorm(A, F, D, n) - 2.f * G11[r];
                int pi = m * (2 * N - m - 1) / 2 + (n - m - 1);
                o[OB + pi] = sqrtf(fmaxf(d2, 0.f));
            }
        }
    }
}

extern "C" void kernel_launch(void* const* d_in, const int* in_sizes, int n_in,
                              void* d_out, int out_size, void* d_ws, size_t ws_size,
                              hipStream_t stream) {
    const float* x = (const float*)d_in[0];
    const int T = in_sizes[0] / (NFEAT * 3);          // 65536
    unsigned int* cnt = (unsigned int*)d_ws;          // [0]=left count, [1]=right count
    int* order = (int*)((char*)d_ws + 16);            // 101 ints
    unsigned char* keepL = (unsigned char*)d_ws + 1024;
    unsigned char* keepR = keepL + T;
    float* out = (float*)d_out;

    init_cnt_kernel<<<1, 64, 0, stream>>>(cnt);
    count_keep_kernel<<<T, 64, 0, stream>>>(x, cnt, keepL, keepR);
    order_kernel<<<1, 1024, 0, stream>>>(cnt, keepL, keepR, order, T);
    feat_kernel<<<OUTROWS, 128, 0, stream>>>(x, cnt, order, out);
}